// MRF_layer_66941360275892
// MI455X (gfx1250) — compile-verified
//
#include <hip/hip_runtime.h>
#include <math.h>
#include <stdint.h>

// Problem constants (match reference)
#define NN 16384
#define K_ITERS 5
// Tiling
#define KC 256          // K-chunk per LDS tile
#define NTILES (NN / KC)
#define LSTR 260        // LDS row stride in floats: %4==0 (b128 alignment) and
                        // 260%64==4 -> a wave's ds_load_b64 pattern covers all
                        // 64 banks exactly once (conflict-free).

typedef __attribute__((ext_vector_type(2))) float v2f;
typedef __attribute__((ext_vector_type(8))) float v8f;

// CDNA5 async global->LDS DMA (ASYNCcnt-tracked); no VGPR round-trip.
__device__ __forceinline__ void async_load_b128(unsigned lds_byte_off,
                                                const float* gaddr) {
  asm volatile("global_load_async_to_lds_b128 %0, %1, off"
               :: "v"(lds_byte_off), "v"(gaddr) : "memory");
}
__device__ __forceinline__ void async_load_b32(unsigned lds_byte_off,
                                               const float* gaddr) {
  asm volatile("global_load_async_to_lds_b32 %0, %1, off"
               :: "v"(lds_byte_off), "v"(gaddr) : "memory");
}
__device__ __forceinline__ void wait_async_le5() {
  asm volatile("s_wait_asynccnt 5" ::: "memory");
}
__device__ __forceinline__ void wait_async_0() {
  asm volatile("s_wait_asynccnt 0" ::: "memory");
}

// ---------------------------------------------------------------------------
// q[i] = pre[i,0];  a[i] = g * (log(pre0+1e-8) - log(pre1+1e-8)),  g=sigmoid(w)
// ---------------------------------------------------------------------------
__global__ __launch_bounds__(256) void init_kernel(const float* __restrict__ pre,
                                                   const float* __restrict__ w,
                                                   float* __restrict__ q,
                                                   float* __restrict__ a) {
  const int i = blockIdx.x * blockDim.x + threadIdx.x;
  if (i >= NN) return;
  const float g  = 1.0f / (1.0f + __expf(-w[0]));
  const float p0 = pre[2 * i];
  const float p1 = pre[2 * i + 1];
  q[i] = p0;
  a[i] = g * (__logf(p0 + 1e-8f) - __logf(p1 + 1e-8f));
}

// ---------------------------------------------------------------------------
// y = s @ q  (and, when COMPUTE_RS, rs = s @ 1) via V_WMMA_F32_16X16X4_F32.
// Block = 16 rows of s, 256 threads = 8 wave32s. Double-buffered pipeline:
// tile t+1 streams HBM->LDS via global_load_async_to_lds_b128 (ASYNCcnt)
// while tile t is consumed by WMMAs fed from conflict-free ds_load_b64.
// ---------------------------------------------------------------------------
template <bool COMPUTE_RS>
__global__ __launch_bounds__(256) void matvec_wmma_kernel(
    const float* __restrict__ s, const float* __restrict__ q,
    float* __restrict__ y, float* __restrict__ rs) {
  __shared__ float tile[2][16 * LSTR];
  __shared__ float qsh[2][KC];
  __shared__ float ysh[16];
  __shared__ float rssh[16];

  const int tid  = threadIdx.x;
  const int n0   = blockIdx.x * 16;   // output rows handled by this block
  const int wv   = tid >> 5;          // wave id (wave32)
  const int lane = tid & 31;
  const int hh   = lane >> 4;         // half-wave select
  const int m    = lane & 15;
  const int kb   = wv * 32;           // this wave's K-slice within a chunk

  if (tid < 16) { ysh[tid] = 0.0f; rssh[tid] = 0.0f; }

  // Per-thread staging geometry: 4 b128 rows + 1 b32 q element per tile.
  const int c  = (tid & 63) << 2;     // column 0..252, step 4
  const int r0 = tid >> 6;            // row group 0..3 (rows r0, r0+4, r0+8, r0+12)
  unsigned ldsT[2][4], ldsQ[2];
  const float* gb[4];
  #pragma unroll
  for (int rr = 0; rr < 4; ++rr) {
    const int r = r0 + 4 * rr;
    ldsT[0][rr] = (unsigned)(uintptr_t)&tile[0][r * LSTR + c];
    ldsT[1][rr] = (unsigned)(uintptr_t)&tile[1][r * LSTR + c];
    gb[rr] = s + (size_t)(n0 + r) * NN + c;
  }
  ldsQ[0] = (unsigned)(uintptr_t)&qsh[0][tid];
  ldsQ[1] = (unsigned)(uintptr_t)&qsh[1][tid];

  // Prologue: stream tile 0 into buffer 0 (5 async ops per wave).
  #pragma unroll
  for (int rr = 0; rr < 4; ++rr) async_load_b128(ldsT[0][rr], gb[rr]);
  async_load_b32(ldsQ[0], q + tid);

  v8f acc  = {};
  v8f accr = {};
  const v2f ones = {1.0f, 1.0f};

  for (int t = 0; t < NTILES; ++t) {
    const int cur = t & 1;
    __syncthreads();  // all reads of the other buffer (iter t-1) are done
    if (t + 1 < NTILES) {
      const int kt1 = (t + 1) * KC;
      const int nxt = cur ^ 1;
      #pragma unroll
      for (int rr = 0; rr < 4; ++rr) async_load_b128(ldsT[nxt][rr], gb[rr] + kt1);
      async_load_b32(ldsQ[nxt], q + kt1 + tid);
      wait_async_le5();  // in-order: the 5 older ops (tile t) have landed
    } else {
      wait_async_0();
    }
    __syncthreads();  // every wave waited on its own ops -> tile t fully visible

    // Preload all fragments for this wave's 32-K slice, then 8 WMMAs of K=4.
    const float* tl = tile[cur];
    const float* qq = qsh[cur];
    v2f af[8], bf[8];
    #pragma unroll
    for (int j = 0; j < 8; ++j) {
      // A (16x4) = q broadcast over rows. Documented layout: lane M=lane%16,
      // VGPR0 = K(2*hh), VGPR1 = K(2*hh+1) -> M-independent fill is exact.
      af[j] = *(const v2f*)(qq + kb + 4 * j + 2 * hh);
      // B (4x16): B[k][n] = s[n0+n][kt+kb+4j+k]; VGPRv = K(v+2*hh), N=lane%16.
      bf[j] = *(const v2f*)(tl + m * LSTR + kb + 4 * j + 2 * hh);
    }
    #pragma unroll
    for (int j = 0; j < 8; ++j) {
      acc = __builtin_amdgcn_wmma_f32_16x16x4_f32(false, af[j], false, bf[j],
                                                  (short)0, acc, false, false);
      if (COMPUTE_RS) {
        accr = __builtin_amdgcn_wmma_f32_16x16x4_f32(false, ones, false, bf[j],
                                                     (short)0, accr, false, false);
      }
    }
  }

  // Every D row equals y[n0..n0+15]; VGPR0 lanes 0..15 hold N=0..15 (M=0).
  if (lane < 16) {
    atomicAdd(&ysh[m], acc[0]);
    if (COMPUTE_RS) atomicAdd(&rssh[m], accr[0]);
  }
  __syncthreads();
  if (tid < 16) {
    y[n0 + tid] = ysh[tid];
    if (COMPUTE_RS) rs[n0 + tid] = rssh[tid];
  }
}

// ---------------------------------------------------------------------------
// q <- sigmoid(a + (1-g) * tanh((2y - rs)/2)); on last iter emit [q, 1-q].
// ---------------------------------------------------------------------------
__global__ __launch_bounds__(256) void update_kernel(
    const float* __restrict__ y, const float* __restrict__ rs,
    const float* __restrict__ a, const float* __restrict__ w,
    float* __restrict__ q, float* __restrict__ out, int last) {
  const int i = blockIdx.x * blockDim.x + threadIdx.x;
  if (i >= NN) return;
  const float g  = 1.0f / (1.0f + __expf(-w[0]));
  const float d  = tanhf(0.5f * (2.0f * y[i] - rs[i]));  // Q2[:,0]-Q2[:,1]
  const float qn = 1.0f / (1.0f + __expf(-(a[i] + (1.0f - g) * d)));
  q[i] = qn;
  if (last) {
    out[2 * i]     = qn;
    out[2 * i + 1] = 1.0f - qn;
  }
}

// ---------------------------------------------------------------------------
extern "C" void kernel_launch(void* const* d_in, const int* in_sizes, int n_in,
                              void* d_out, int out_size, void* d_ws, size_t ws_size,
                              hipStream_t stream) {
  (void)in_sizes; (void)n_in; (void)out_size; (void)ws_size;
  const float* pre = (const float*)d_in[0];   // [N,2]
  const float* s   = (const float*)d_in[1];   // [N,N]
  const float* w   = (const float*)d_in[2];   // [1]
  float* out = (float*)d_out;                 // [N,2]

  float* q  = (float*)d_ws;                   // [N]
  float* a  = q + NN;                         // [N]
  float* y  = q + 2 * NN;                     // [N]
  float* rs = q + 3 * NN;                     // [N]

  const dim3 eb(256), eg(NN / 256);
  init_kernel<<<eg, eb, 0, stream>>>(pre, w, q, a);

  const dim3 mb(256), mg(NN / 16);
  for (int it = 0; it < K_ITERS; ++it) {
    if (it == 0) {
      matvec_wmma_kernel<true><<<mg, mb, 0, stream>>>(s, q, y, rs);
    } else {
      matvec_wmma_kernel<false><<<mg, mb, 0, stream>>>(s, q, y, rs);
    }
    update_kernel<<<eg, eb, 0, stream>>>(y, rs, a, w, q, out,
                                         it == K_ITERS - 1 ? 1 : 0);
  }
}